// dumb_GNN_44813688766467
// MI455X (gfx1250) — compile-verified
//
#include <hip/hip_runtime.h>
#include <math.h>

typedef __attribute__((ext_vector_type(2))) float v2f;
typedef __attribute__((ext_vector_type(8))) float v8f;

#define IN_DIM 768
#define HID 256

// ---------------------------------------------------------------------------
// 1) zero int buffer (used for counts, then reused as fill cursors)
// ---------------------------------------------------------------------------
__global__ void zero_int_kernel(int* __restrict__ p, int n) {
  int i = blockIdx.x * blockDim.x + threadIdx.x;
  if (i < n) p[i] = 0;
}

// 2) in-degree counts over dst (int atomics, E ops total)
__global__ void count_deg_kernel(const int* __restrict__ dst,
                                 int* __restrict__ counts, int e) {
  int i = blockIdx.x * blockDim.x + threadIdx.x;
  if (i < e) atomicAdd(&counts[dst[i]], 1);
}

// 3) exclusive scan counts -> rowptr[0..n]  (single block, chunked H-S scan)
__global__ __launch_bounds__(1024) void scan_kernel(
    const int* __restrict__ counts, int* __restrict__ rowptr, int n) {
  __shared__ int smem[1024];
  __shared__ int running;
  const int tid = threadIdx.x;
  if (tid == 0) running = 0;
  __syncthreads();
  for (int base = 0; base < n; base += 1024) {
    int i = base + tid;
    int v = (i < n) ? counts[i] : 0;
    smem[tid] = v;
    __syncthreads();
#pragma unroll
    for (int off = 1; off < 1024; off <<= 1) {
      int t = (tid >= off) ? smem[tid - off] : 0;
      __syncthreads();
      smem[tid] += t;
      __syncthreads();
    }
    if (i < n) rowptr[i] = running + smem[tid] - v;  // exclusive
    __syncthreads();
    if (tid == 0) running += smem[1023];
    __syncthreads();
  }
  if (tid == 0) rowptr[n] = running;
}

// 4) dinv[i] = rsqrt(deg) with deg = in-degree + 1 (self loop); always >= 1
__global__ void dinv_kernel(const int* __restrict__ counts,
                            float* __restrict__ dinv, int n) {
  int i = blockIdx.x * blockDim.x + threadIdx.x;
  if (i < n) dinv[i] = rsqrtf((float)(counts[i] + 1));
}

// 5) bin edges into CSR: col[rowptr[dst] + cursor[dst]++] = src
__global__ void fill_kernel(const int* __restrict__ src,
                            const int* __restrict__ dst,
                            const int* __restrict__ rowptr,
                            int* __restrict__ cursor, int* __restrict__ col,
                            int e) {
  int i = blockIdx.x * blockDim.x + threadIdx.x;
  if (i < e) {
    int d = dst[i];
    int pos = rowptr[d] + atomicAdd(&cursor[d], 1);
    col[pos] = src[i];
  }
}

// ---------------------------------------------------------------------------
// 6) H = X @ W1   [N,768] x [768,256] -> [N,256]
//    fp32 WMMA 16x16x4. Block = 256 thr = 8 waves, owns 16 rows x 256 cols.
//    A strip staged in LDS, row stride 772 dwords (16B-aligned, conflict-free)
// ---------------------------------------------------------------------------
#define A_STRIDE 772

__global__ __launch_bounds__(256) void gemm1_wmma_kernel(
    const float* __restrict__ X, const float* __restrict__ W1,
    float* __restrict__ H) {
  __shared__ float Alds[16 * A_STRIDE];

  const int tid = threadIdx.x;
  const int mBase = blockIdx.x * 16;

  for (int i = tid; i < 16 * (IN_DIM / 4); i += 256) {
    int r = i / (IN_DIM / 4);
    int c = (i % (IN_DIM / 4)) * 4;
    float4 v = *(const float4*)(X + (size_t)(mBase + r) * IN_DIM + c);
    *(float4*)(&Alds[r * A_STRIDE + c]) = v;
  }
  __syncthreads();

  const int lane = tid & 31;
  const int wave = tid >> 5;
  const int halfSel = lane >> 4;  // 0: K pair {0,1}   1: K pair {2,3}
  const int idx = lane & 15;      // M for A-frag, N for B-frag
  const int nBase0 = (wave * 2) * 16;
  const int nBase1 = nBase0 + 16;

  v8f acc0 = {};
  v8f acc1 = {};

#pragma unroll 2
  for (int k0 = 0; k0 < IN_DIM; k0 += 4) {
    const int kk = k0 + 2 * halfSel;
    v2f a = *(const v2f*)&Alds[idx * A_STRIDE + kk];
    v2f b0, b1;
    b0.x = W1[(size_t)kk * HID + nBase0 + idx];
    b0.y = W1[(size_t)(kk + 1) * HID + nBase0 + idx];
    b1.x = W1[(size_t)kk * HID + nBase1 + idx];
    b1.y = W1[(size_t)(kk + 1) * HID + nBase1 + idx];
    acc0 = __builtin_amdgcn_wmma_f32_16x16x4_f32(false, a, false, b0,
                                                 (short)0, acc0, false, false);
    acc1 = __builtin_amdgcn_wmma_f32_16x16x4_f32(false, a, false, b1,
                                                 (short)0, acc1, false, false);
  }

  const int mOff = halfSel * 8;
  float* Hrow = H + (size_t)mBase * HID;
#pragma unroll
  for (int i = 0; i < 8; ++i) {
    Hrow[(size_t)(mOff + i) * HID + nBase0 + idx] = acc0[i];
    Hrow[(size_t)(mOff + i) * HID + nBase1 + idx] = acc1[i];
  }
}

// ---------------------------------------------------------------------------
// 7) fused: CSR gather-aggregate + self-loop + bias + relu + 256x3 GEMV
//    + log_softmax.  One wave per node; lane owns features {4*lane..4*lane+3}
//    and {128+4*lane..+3}  ->  2 float4 accumulators, coalesced 512B rows.
//    No fp atomics anywhere; H is L2-resident (102 MB < 192 MB L2).
// ---------------------------------------------------------------------------
__global__ __launch_bounds__(256) void gather_head_kernel(
    const int* __restrict__ rowptr, const int* __restrict__ col,
    const float* __restrict__ dinv, const float* __restrict__ H,
    const float* __restrict__ b1, const float* __restrict__ W2,
    const float* __restrict__ b2, float* __restrict__ out, int n) {
  __shared__ float w2s[3 * HID];  // transposed: w2s[c*256 + j] = W2[j][c]
  __shared__ float b1s[HID];
  const int tid = threadIdx.x;
  for (int i = tid; i < 3 * HID; i += 256) {
    int c = i >> 8;
    int j = i & 255;
    w2s[c * HID + j] = W2[(size_t)j * 3 + c];
  }
  for (int i = tid; i < HID; i += 256) b1s[i] = b1[i];
  __syncthreads();

  const int lane = tid & 31;
  const int node = blockIdx.x * 8 + (tid >> 5);
  if (node >= n) return;

  const int f0 = lane * 4;
  const int f1 = 128 + lane * 4;
  const float dd = dinv[node];
  const float sw = dd * dd;  // self-loop norm

  const float* hn = H + (size_t)node * HID;
  float4 h0 = *(const float4*)(hn + f0);
  float4 h1 = *(const float4*)(hn + f1);
  float a00 = h0.x * sw + b1s[f0 + 0];
  float a01 = h0.y * sw + b1s[f0 + 1];
  float a02 = h0.z * sw + b1s[f0 + 2];
  float a03 = h0.w * sw + b1s[f0 + 3];
  float a10 = h1.x * sw + b1s[f1 + 0];
  float a11 = h1.y * sw + b1s[f1 + 1];
  float a12 = h1.z * sw + b1s[f1 + 2];
  float a13 = h1.w * sw + b1s[f1 + 3];

  const int beg = rowptr[node];
  const int end = rowptr[node + 1];
  for (int k = beg; k < end; ++k) {
    int s = col[k];  // wave-uniform
    float w = dinv[s] * dd;
    const float* hs = H + (size_t)s * HID;
    float4 g0 = *(const float4*)(hs + f0);
    float4 g1 = *(const float4*)(hs + f1);
    a00 += g0.x * w; a01 += g0.y * w; a02 += g0.z * w; a03 += g0.w * w;
    a10 += g1.x * w; a11 += g1.y * w; a12 += g1.z * w; a13 += g1.w * w;
  }

  // relu + 3-class GEMV (per-lane partials over this lane's 8 features)
  float c0 = 0.f, c1 = 0.f, c2 = 0.f;
  {
    float v;
    v = fmaxf(a00, 0.f); c0 += v * w2s[f0 + 0]; c1 += v * w2s[HID + f0 + 0]; c2 += v * w2s[2 * HID + f0 + 0];
    v = fmaxf(a01, 0.f); c0 += v * w2s[f0 + 1]; c1 += v * w2s[HID + f0 + 1]; c2 += v * w2s[2 * HID + f0 + 1];
    v = fmaxf(a02, 0.f); c0 += v * w2s[f0 + 2]; c1 += v * w2s[HID + f0 + 2]; c2 += v * w2s[2 * HID + f0 + 2];
    v = fmaxf(a03, 0.f); c0 += v * w2s[f0 + 3]; c1 += v * w2s[HID + f0 + 3]; c2 += v * w2s[2 * HID + f0 + 3];
    v = fmaxf(a10, 0.f); c0 += v * w2s[f1 + 0]; c1 += v * w2s[HID + f1 + 0]; c2 += v * w2s[2 * HID + f1 + 0];
    v = fmaxf(a11, 0.f); c0 += v * w2s[f1 + 1]; c1 += v * w2s[HID + f1 + 1]; c2 += v * w2s[2 * HID + f1 + 1];
    v = fmaxf(a12, 0.f); c0 += v * w2s[f1 + 2]; c1 += v * w2s[HID + f1 + 2]; c2 += v * w2s[2 * HID + f1 + 2];
    v = fmaxf(a13, 0.f); c0 += v * w2s[f1 + 3]; c1 += v * w2s[HID + f1 + 3]; c2 += v * w2s[2 * HID + f1 + 3];
  }
#pragma unroll
  for (int off = 16; off >= 1; off >>= 1) {
    c0 += __shfl_down(c0, off, 32);
    c1 += __shfl_down(c1, off, 32);
    c2 += __shfl_down(c2, off, 32);
  }
  if (lane == 0) {
    float l0 = c0 + b2[0], l1 = c1 + b2[1], l2 = c2 + b2[2];
    float m = fmaxf(l0, fmaxf(l1, l2));
    float lse = m + logf(expf(l0 - m) + expf(l1 - m) + expf(l2 - m));
    float* o = out + (size_t)node * 3;
    o[0] = l0 - lse;
    o[1] = l1 - lse;
    o[2] = l2 - lse;
  }
}

// ---------------------------------------------------------------------------
extern "C" void kernel_launch(void* const* d_in, const int* in_sizes, int n_in,
                              void* d_out, int out_size, void* d_ws,
                              size_t ws_size, hipStream_t stream) {
  const float* X  = (const float*)d_in[0];
  const int*   EI = (const int*)d_in[1];
  const float* W1 = (const float*)d_in[2];
  const float* b1 = (const float*)d_in[3];
  const float* W2 = (const float*)d_in[4];
  const float* b2 = (const float*)d_in[5];

  const int N = in_sizes[0] / IN_DIM;  // 100000
  const int E = in_sizes[1] / 2;       // 1600000
  const int* src = EI;
  const int* dst = EI + E;

  // workspace layout: counts | rowptr | dinv | col | H  (all 256B aligned)
  char* base = (char*)d_ws;
  size_t o = 0;
  auto take = [&](size_t bytes) {
    char* p = base + o;
    o = (o + bytes + 255) & ~(size_t)255;
    return p;
  };
  int*   counts = (int*)take((size_t)N * 4);
  int*   rowptr = (int*)take(((size_t)N + 1) * 4);
  float* dinv   = (float*)take((size_t)N * 4);
  int*   col    = (int*)take((size_t)E * 4);
  float* H      = (float*)take((size_t)N * HID * 4);

  // CSR build
  zero_int_kernel<<<(N + 255) / 256, 256, 0, stream>>>(counts, N);
  count_deg_kernel<<<(E + 255) / 256, 256, 0, stream>>>(dst, counts, E);
  scan_kernel<<<1, 1024, 0, stream>>>(counts, rowptr, N);
  dinv_kernel<<<(N + 255) / 256, 256, 0, stream>>>(counts, dinv, N);
  zero_int_kernel<<<(N + 255) / 256, 256, 0, stream>>>(counts, N);  // cursors
  fill_kernel<<<(E + 255) / 256, 256, 0, stream>>>(src, dst, rowptr, counts,
                                                   col, E);

  // dense transform (WMMA)
  gemm1_wmma_kernel<<<N / 16, 256, 0, stream>>>(X, W1, H);  // N % 16 == 0

  // fused aggregate + relu + linear + log_softmax
  gather_head_kernel<<<(N + 7) / 8, 256, 0, stream>>>(rowptr, col, dinv, H, b1,
                                                      W2, b2, (float*)d_out, N);
}